// TripletLoss_80668075754103
// MI455X (gfx1250) — compile-verified
//
#include <hip/hip_runtime.h>
#include <hip/hip_bf16.h>

typedef __attribute__((ext_vector_type(2))) float v2f;
typedef __attribute__((ext_vector_type(8))) float v8f;

#define N_EMB 8192
#define DIM   128
#define KSTEPS (DIM / 4)        // 32 WMMA k-steps of 4
#define TILES  (N_EMB / 16)     // 512 column tiles
#define BIGF   1000000.0f

// ---------------------------------------------------------------------------
// Kernel 1: squared L2 norms per row.  sq[i] = sum_k X[i][k]^2
// ---------------------------------------------------------------------------
__global__ void k_sqnorm(const float* __restrict__ X, float* __restrict__ sq) {
    int i = blockIdx.x * blockDim.x + threadIdx.x;
    if (i >= N_EMB) return;
    const float4* row = (const float4*)(X + (size_t)i * DIM);
    float s = 0.0f;
#pragma unroll
    for (int k = 0; k < DIM / 4; ++k) {
        float4 v = row[k];
        s += v.x * v.x + v.y * v.y + v.z * v.z + v.w * v.w;
    }
    sq[i] = s;
}

// ---------------------------------------------------------------------------
// Kernel 2: fused Gram (fp32 WMMA) + distance + batch-hard mining.
// One wave32 per 16-row strip; 4 waves (128 threads) per block.
// For V_WMMA_F32_16X16X4_F32, lane L (half = L>>4, l16 = L&15):
//   A frag (16x4):  a.x = A[l16][4*ks + 2*half],  a.y = next k   (contig pair)
//   B frag (4x16):  b.x = X[col(l16)][4*ks + 2*half], b.y = next (contig pair)
//   C (16x16):      VGPR v -> row (v + 8*half), col l16
// ---------------------------------------------------------------------------
__global__ void __launch_bounds__(128)
k_triplet(const float* __restrict__ X, const int* __restrict__ labels,
          const float* __restrict__ sq, const float* __restrict__ marginp,
          float* __restrict__ loss) {
    const int lane = threadIdx.x & 31;
    const int wave = threadIdx.x >> 5;
    const int half = lane >> 4;   // 0: lanes 0-15, 1: lanes 16-31
    const int l16  = lane & 15;
    const int rowBase = (blockIdx.x * 4 + wave) * 16;
    const float margin = marginp[0];

    // ---- Preload this strip's A fragments: 32 contiguous float2 per lane ----
    const int rowA = rowBase + l16;
    const float* arow = X + (size_t)rowA * DIM + 2 * half;
    v2f a[KSTEPS];
#pragma unroll
    for (int ks = 0; ks < KSTEPS; ++ks)
        a[ks] = *(const v2f*)(arow + 4 * ks);

    // ---- Per-lane row metadata matching the C-matrix layout ----
    float sqRow[8];
    int   labRow[8];
    int   rowIdx[8];
#pragma unroll
    for (int v = 0; v < 8; ++v) {
        int r = rowBase + v + 8 * half;
        rowIdx[v] = r;
        sqRow[v]  = sq[r];
        labRow[v] = labels[r];
    }

    float posm[8], negm[8];
#pragma unroll
    for (int v = 0; v < 8; ++v) { posm[v] = 0.0f; negm[v] = 3.0e38f; }

    // ---- Stream over all 512 column tiles ----
    for (int t = 0; t < TILES; ++t) {
        const int col = t * 16 + l16;
        const float* bcol = X + (size_t)col * DIM + 2 * half;
        if (t + 1 < TILES)
            __builtin_prefetch(bcol + 16 * DIM, 0, 3);   // global_prefetch_b8

        v8f c0 = {0.f, 0.f, 0.f, 0.f, 0.f, 0.f, 0.f, 0.f};
        v8f c1 = {0.f, 0.f, 0.f, 0.f, 0.f, 0.f, 0.f, 0.f};
#pragma unroll
        for (int ks = 0; ks < KSTEPS; ks += 2) {
            v2f b0 = *(const v2f*)(bcol + 4 * ks);
            v2f b1 = *(const v2f*)(bcol + 4 * (ks + 1));
            c0 = __builtin_amdgcn_wmma_f32_16x16x4_f32(
                    false, a[ks],     false, b0, (short)0, c0, false, false);
            c1 = __builtin_amdgcn_wmma_f32_16x16x4_f32(
                    false, a[ks + 1], false, b1, (short)0, c1, false, false);
        }

        const float sqC  = sq[col];
        const int   labC = labels[col];
#pragma unroll
        for (int v = 0; v < 8; ++v) {
            float dot = c0[v] + c1[v];
            float d2  = fmaxf(sqRow[v] + sqC - 2.0f * dot, 0.0f);
            float d   = __builtin_sqrtf(d2);
            bool same = (labC == labRow[v]);
            bool self = (col == rowIdx[v]);
            float pc  = (same && !self) ? d : 0.0f;        // dist * mask_pos
            posm[v] = fmaxf(posm[v], pc);
            float nc  = d + (same ? BIGF : 0.0f);          // dist + (1-mask_neg)*BIG
            negm[v] = fminf(negm[v], nc);
        }
    }

    // ---- Butterfly reduce across the 16 lanes holding the same rows ----
#pragma unroll
    for (int v = 0; v < 8; ++v) {
        float p = posm[v], n = negm[v];
#pragma unroll
        for (int m = 8; m >= 1; m >>= 1) {
            p = fmaxf(p, __shfl_xor(p, m));
            n = fminf(n, __shfl_xor(n, m));
        }
        if (l16 == 0)
            loss[rowIdx[v]] = fmaxf(p - n + margin, 0.0f);
    }
}

// ---------------------------------------------------------------------------
// Kernel 3: deterministic mean over the 8192 per-row losses.
// ---------------------------------------------------------------------------
__global__ void k_reduce(const float* __restrict__ loss, float* __restrict__ out) {
    __shared__ float smem[256];
    float acc = 0.0f;
    for (int i = threadIdx.x; i < N_EMB; i += 256)
        acc += loss[i];
    smem[threadIdx.x] = acc;
    __syncthreads();
#pragma unroll
    for (int s = 128; s > 0; s >>= 1) {
        if (threadIdx.x < s) smem[threadIdx.x] += smem[threadIdx.x + s];
        __syncthreads();
    }
    if (threadIdx.x == 0) out[0] = smem[0] / (float)N_EMB;
}

// ---------------------------------------------------------------------------
extern "C" void kernel_launch(void* const* d_in, const int* in_sizes, int n_in,
                              void* d_out, int out_size, void* d_ws, size_t ws_size,
                              hipStream_t stream) {
    const float* X       = (const float*)d_in[0];   // [8192,128] f32
    const int*   labels  = (const int*)d_in[1];     // [8192] int
    const float* margin  = (const float*)d_in[2];   // scalar f32
    float*       out     = (float*)d_out;

    float* sq   = (float*)d_ws;                     // 8192 f32
    float* loss = sq + N_EMB;                       // 8192 f32

    k_sqnorm <<<N_EMB / 256, 256, 0, stream>>>(X, sq);
    k_triplet<<<N_EMB / 64,  128, 0, stream>>>(X, labels, sq, margin, loss);
    k_reduce <<<1,           256, 0, stream>>>(loss, out);
}